// CompLinear2_24876450578671
// MI455X (gfx1250) — compile-verified
//
#include <hip/hip_runtime.h>
#include <hip/hip_bf16.h>

typedef __attribute__((ext_vector_type(2))) float v2f;
typedef __attribute__((ext_vector_type(8))) float v8f;

#define IN_F  8192
#define OUT_F 8192
#define BATCH 16

// ---------------------------------------------------------------------------
// Kernel 1: per-batch-row sums of input  ->  ws[0..15]
// ---------------------------------------------------------------------------
__global__ __launch_bounds__(256) void rowsum_kernel(const float* __restrict__ input,
                                                     float* __restrict__ ws) {
    __shared__ float s[256];
    const int b = blockIdx.x;
    float acc = 0.0f;
    for (int i = threadIdx.x; i < IN_F; i += 256)
        acc += input[b * IN_F + i];
    s[threadIdx.x] = acc;
    __syncthreads();
    for (int off = 128; off > 0; off >>= 1) {
        if ((int)threadIdx.x < off) s[threadIdx.x] += s[threadIdx.x + off];
        __syncthreads();
    }
    if (threadIdx.x == 0) ws[b] = s[0];
}

// ---------------------------------------------------------------------------
// Kernel 2: bandwidth-bound WMMA GEMM with fused de-standardization epilogue.
//   block = 128 threads = 4 waves, one 16x16 output tile per block,
//   K split 4 ways across waves, LDS reduction, wave0 does epilogue.
// ---------------------------------------------------------------------------
__global__ __launch_bounds__(128) void comp_linear_kernel(
        const float* __restrict__ input,   // [16, 8192]
        const float* __restrict__ hatWr,   // [8192, 8192]
        const float* __restrict__ scale,   // [8192]
        const float* __restrict__ mean,    // [8192]
        const float* __restrict__ bias,    // [8192]
        const float* __restrict__ rowsum,  // [16]  (from kernel 1)
        float* __restrict__ out)           // [16, 8192]
{
    __shared__ float red[3 * 256];  // partials from waves 1..3

    const int lane  = threadIdx.x & 31;
    const int wave  = threadIdx.x >> 5;
    const int nbase = blockIdx.x * 16;      // 512 tiles over OUT_F

    // f32 WMMA fragment layout (16x4 A / 4x16 B, wave32):
    //   lane l: row/col index = l & 15,  K-pair offset = (l>>4)*2
    const int rc   = lane & 15;
    const int koff = (lane >> 4) * 2;

    const int kPerWave = IN_F / 4;          // 2048
    const int kbeg     = wave * kPerWave;

    const float* aPtr = input + (size_t)rc * IN_F + kbeg + koff;
    const float* bPtr = hatWr + ((size_t)(nbase + rc)) * IN_F + kbeg + koff;

    v8f acc0 = {};
    v8f acc1 = {};

    // 512 WMMA K=4 steps per wave; unroll 4 -> 8 b64 loads in flight / iter
    for (int k = 0; k < kPerWave; k += 16) {
        v2f a0 = *(const v2f*)(aPtr + k);
        v2f b0 = *(const v2f*)(bPtr + k);
        v2f a1 = *(const v2f*)(aPtr + k + 4);
        v2f b1 = *(const v2f*)(bPtr + k + 4);
        v2f a2 = *(const v2f*)(aPtr + k + 8);
        v2f b2 = *(const v2f*)(bPtr + k + 8);
        v2f a3 = *(const v2f*)(aPtr + k + 12);
        v2f b3 = *(const v2f*)(bPtr + k + 12);

        acc0 = __builtin_amdgcn_wmma_f32_16x16x4_f32(false, a0, false, b0,
                                                     (short)0, acc0, false, false);
        acc1 = __builtin_amdgcn_wmma_f32_16x16x4_f32(false, a1, false, b1,
                                                     (short)0, acc1, false, false);
        acc0 = __builtin_amdgcn_wmma_f32_16x16x4_f32(false, a2, false, b2,
                                                     (short)0, acc0, false, false);
        acc1 = __builtin_amdgcn_wmma_f32_16x16x4_f32(false, a3, false, b3,
                                                     (short)0, acc1, false, false);
    }

    v8f acc = acc0 + acc1;

    // Cross-wave K reduction through LDS
    if (wave != 0) {
        float* dst = &red[(wave - 1) * 256];
#pragma unroll
        for (int r = 0; r < 8; ++r)
            dst[r * 32 + lane] = acc[r];
    }
    __syncthreads();

    if (wave == 0) {
        // C/D layout: lane l, VGPR r ->  M = r + (l>>4)*8 , N = l&15
        const int N = lane & 15;
        const int o = nbase + N;
        const float sc = scale[o];
        const float mn = mean[o];
        const float bi = bias[o];
#pragma unroll
        for (int r = 0; r < 8; ++r) {
            float v = acc[r] + red[0 * 256 + r * 32 + lane]
                             + red[1 * 256 + r * 32 + lane]
                             + red[2 * 256 + r * 32 + lane];
            const int M = r + (lane >> 4) * 8;
            out[(size_t)M * OUT_F + o] = sc * v + mn * rowsum[M] + bi;
        }
    }
}

// ---------------------------------------------------------------------------
extern "C" void kernel_launch(void* const* d_in, const int* in_sizes, int n_in,
                              void* d_out, int out_size, void* d_ws, size_t ws_size,
                              hipStream_t stream) {
    const float* input = (const float*)d_in[0];
    const float* hatWr = (const float*)d_in[1];
    const float* scale = (const float*)d_in[2];
    const float* mean  = (const float*)d_in[3];
    const float* bias  = (const float*)d_in[4];
    float* out = (float*)d_out;
    float* ws  = (float*)d_ws;   // 16 floats of scratch for row sums

    rowsum_kernel<<<BATCH, 256, 0, stream>>>(input, ws);

    const int nTiles = OUT_F / 16;  // 512 blocks, 4 waves each
    comp_linear_kernel<<<nTiles, 128, 0, stream>>>(input, hatWr, scale, mean,
                                                   bias, ws, out);
}